// ClassificationHead_10084583211447
// MI455X (gfx1250) — compile-verified
//
#include <hip/hip_runtime.h>
#include <hip/hip_bf16.h>

// ---------------- problem constants (match reference) ----------------
constexpr int NW   = 5;       // n_way
constexpr int NS   = 125;     // n_support
constexpr int NQ   = 2048;    // n_query
constexpr int DIM  = 8192;    // feature dim
constexpr int NZ   = NW * NS; // 625  (QP variable count)
constexpr int NP   = NS;      // 125  (equality constraints)
constexpr int NT   = NZ + NP; // 750  (KKT size)
constexpr int KKTS = 768;     // KKT row stride (padded)
constexpr float C_REG = 0.1f;
constexpr float SIGMA = 0.1f;
constexpr int MAX_ITER = 20;

// ---------------- workspace layout (floats) ----------------
constexpr size_t OFF_K      = 0;                       // 128x128
constexpr size_t OFF_COMPAT = OFF_K + 128 * 128;       // 128x2048
constexpr size_t OFF_Z      = OFF_COMPAT + 128 * 2048; // 640
constexpr size_t OFF_S      = OFF_Z + 640;             // 640
constexpr size_t OFF_LAM    = OFF_S + 640;             // 640
constexpr size_t OFF_NU     = OFF_LAM + 640;           // 128
constexpr size_t OFF_R2     = OFF_NU + 128;            // 640
constexpr size_t OFF_R3     = OFF_R2 + 640;            // 640
constexpr size_t OFF_RHS    = OFF_R3 + 640;            // 768
constexpr size_t OFF_SOL    = OFF_RHS + 768;           // 768
constexpr size_t OFF_DS     = OFF_SOL + 768;           // 640
constexpr size_t OFF_DLAM   = OFF_DS + 640;            // 640
constexpr size_t OFF_KKT    = OFF_DLAM + 640;          // 768x768

typedef float v2f __attribute__((ext_vector_type(2)));
typedef float v8f __attribute__((ext_vector_type(8)));

__device__ __forceinline__ int imin(int a, int b) { return a < b ? a : b; }

// ======================================================================
// Kernel 1: K = S * S^T  (125x125, padded to 128x128), f32 WMMA 16x16x4.
// One wave per 16x16 tile; grid (8,8). Branch-free: out-of-range rows are
// clamped to row 124 — polluted padding (rows/cols 125..127) is never read.
// Per 8-wide k-chunk: one float4 load per matrix feeds two WMMAs
// (K-permutation: half 0 lanes cover k0..k0+3, half 1 lanes k0+4..k0+7).
// ======================================================================
__global__ void __launch_bounds__(32) kmat_wmma(const float* __restrict__ S,
                                                float* __restrict__ ws) {
  float* Kout = ws + OFF_K;
  const int lane = threadIdx.x;
  const int m0 = blockIdx.y * 16;
  const int n0 = blockIdx.x * 16;
  const int r    = lane & 15;
  const int half = lane >> 4;

  const int mr = imin(m0 + r, NS - 1);
  const int nr = imin(n0 + r, NS - 1);
  const float* Arow = S + (size_t)mr * DIM;
  const float* Brow = S + (size_t)nr * DIM;

  v8f acc = {};
  for (int k0 = 0; k0 < DIM; k0 += 8) {
    if ((k0 & 511) == 0 && k0 + 512 < DIM) {
      __builtin_prefetch(Arow + k0 + 512, 0, 1);   // global_prefetch_b8
      __builtin_prefetch(Brow + k0 + 512, 0, 1);
    }
    const float4 ta = *(const float4*)(Arow + k0 + half * 4);
    const float4 tb = *(const float4*)(Brow + k0 + half * 4);
    v2f a1 = {ta.x, ta.y}, b1 = {tb.x, tb.y};
    v2f a2 = {ta.z, ta.w}, b2 = {tb.z, tb.w};
    acc = __builtin_amdgcn_wmma_f32_16x16x4_f32(false, a1, false, b1, (short)0, acc, false, false);
    acc = __builtin_amdgcn_wmma_f32_16x16x4_f32(false, a2, false, b2, (short)0, acc, false, false);
  }

  const int row_lo = (lane < 16) ? 0 : 8;
  for (int rr = 0; rr < 8; ++rr)
    Kout[(size_t)(m0 + row_lo + rr) * 128 + n0 + r] = acc[rr];
}

// ======================================================================
// Kernel 2: compat = S * Q^T  (125x2048 padded to 128 rows), f32 WMMA with
// async global->LDS double buffering (the dominant 4.2-GFLOP GEMM).
// Each 16-wide k-chunk of the A (support) and B (query) tiles is staged
// into LDS with global_load_async_to_lds_b128 (ASYNCcnt) while the current
// chunk is consumed from LDS (ds_load_b128) by 4 WMMAs. One wave/block, so
// s_wait_asynccnt alone fences the buffer (async completions are in-order).
// ======================================================================
__global__ void __launch_bounds__(32) compat_wmma(const float* __restrict__ S,
                                                  const float* __restrict__ Q,
                                                  float* __restrict__ ws) {
  __shared__ float shA[2][256];   // 16 rows x 16 k  per buffer
  __shared__ float shB[2][256];

  float* C = ws + OFF_COMPAT;
  const int lane = threadIdx.x;
  const int s0 = blockIdx.y * 16;
  const int q0 = blockIdx.x * 16;
  const int r    = lane & 15;
  const int half = lane >> 4;

  // Staging granules: 64 float4s per 16x16 tile; lane stages granule
  // g0 = lane (rows 0..7) and g1 = lane+32 (rows 8..15).
  const int grow = lane >> 2;            // 0..7
  const int gk   = (lane & 3) * 4;       // k-quad within chunk
  const float* Ar0 = S + (size_t)imin(s0 + grow,     NS - 1) * DIM + gk;
  const float* Ar1 = S + (size_t)imin(s0 + grow + 8, NS - 1) * DIM + gk;
  const float* Br0 = Q + (size_t)(q0 + grow)     * DIM + gk;
  const float* Br1 = Q + (size_t)(q0 + grow + 8) * DIM + gk;

  unsigned aoff[2], boff[2];
  for (int buf = 0; buf < 2; ++buf) {
    aoff[buf] = (unsigned)(size_t)(&shA[buf][grow * 16 + gk]);
    boff[buf] = (unsigned)(size_t)(&shB[buf][grow * 16 + gk]);
  }

  auto stage = [&](int buf, int k0) {
    asm volatile("global_load_async_to_lds_b128 %0, %1, off"
                 :: "v"(aoff[buf]),       "v"(Ar0 + k0) : "memory");
    asm volatile("global_load_async_to_lds_b128 %0, %1, off"
                 :: "v"(aoff[buf] + 512u), "v"(Ar1 + k0) : "memory");
    asm volatile("global_load_async_to_lds_b128 %0, %1, off"
                 :: "v"(boff[buf]),       "v"(Br0 + k0) : "memory");
    asm volatile("global_load_async_to_lds_b128 %0, %1, off"
                 :: "v"(boff[buf] + 512u), "v"(Br1 + k0) : "memory");
  };

  v8f acc = {};
  stage(0, 0);
  int buf = 0;
  for (int k0 = 0; k0 < DIM; k0 += 16) {
    const int nk = k0 + 16;
    if (nk < DIM) {
      stage(buf ^ 1, nk);
      asm volatile("s_wait_asynccnt 0x4" ::: "memory"); // current buf done, next in flight
    } else {
      asm volatile("s_wait_asynccnt 0x0" ::: "memory");
    }
    // consume buffer: lane reads its float4s (ds_load_b128)
    const float4 a1 = *(const float4*)(&shA[buf][r * 16 + half * 4]);
    const float4 a2 = *(const float4*)(&shA[buf][r * 16 + 8 + half * 4]);
    const float4 b1 = *(const float4*)(&shB[buf][r * 16 + half * 4]);
    const float4 b2 = *(const float4*)(&shB[buf][r * 16 + 8 + half * 4]);
    v2f pa, pb;
    pa = (v2f){a1.x, a1.y}; pb = (v2f){b1.x, b1.y};
    acc = __builtin_amdgcn_wmma_f32_16x16x4_f32(false, pa, false, pb, (short)0, acc, false, false);
    pa = (v2f){a1.z, a1.w}; pb = (v2f){b1.z, b1.w};
    acc = __builtin_amdgcn_wmma_f32_16x16x4_f32(false, pa, false, pb, (short)0, acc, false, false);
    pa = (v2f){a2.x, a2.y}; pb = (v2f){b2.x, b2.y};
    acc = __builtin_amdgcn_wmma_f32_16x16x4_f32(false, pa, false, pb, (short)0, acc, false, false);
    pa = (v2f){a2.z, a2.w}; pb = (v2f){b2.z, b2.w};
    acc = __builtin_amdgcn_wmma_f32_16x16x4_f32(false, pa, false, pb, (short)0, acc, false, false);
    buf ^= 1;
  }

  const int row_lo = (lane < 16) ? 0 : 8;
  for (int rr = 0; rr < 8; ++rr)
    C[(size_t)(s0 + row_lo + rr) * NQ + q0 + r] = acc[rr];
}

// ======================================================================
// Kernel 3: interior-point QP solve (single workgroup, 256 threads).
// Mirrors the reference: builds the 750x750 KKT system each iteration and
// solves with partially-pivoted Gaussian elimination. G = kron(K,I5)+I is
// never materialized.
// ======================================================================
__global__ void __launch_bounds__(256) qp_solve_kernel(const int* __restrict__ labels,
                                                       float* __restrict__ ws) {
  const int tid = threadIdx.x;
  const int NTH = 256;

  float* Km   = ws + OFF_K;
  float* z    = ws + OFF_Z;
  float* sv   = ws + OFF_S;
  float* lam  = ws + OFF_LAM;
  float* nu   = ws + OFF_NU;
  float* r2   = ws + OFF_R2;
  float* r3   = ws + OFF_R3;
  float* rhs  = ws + OFF_RHS;
  float* sol  = ws + OFF_SOL;
  float* dsv  = ws + OFF_DS;
  float* dlam = ws + OFF_DLAM;
  float* KKT  = ws + OFF_KKT;

  __shared__ float red[256];
  __shared__ int s_piv;

  for (int i = tid; i < NZ; i += NTH) { z[i] = 0.f; sv[i] = 1.f; lam[i] = 1.f; }
  for (int i = tid; i < NP; i += NTH) nu[i] = 0.f;
  __syncthreads();

  for (int iter = 0; iter < MAX_ITER; ++iter) {
    // ---- mu = SIGMA * sum(s*lam) / NZ ----
    float part = 0.f;
    for (int i = tid; i < NZ; i += NTH) part += sv[i] * lam[i];
    red[tid] = part; __syncthreads();
    for (int st = 128; st > 0; st >>= 1) { if (tid < st) red[tid] += red[tid + st]; __syncthreads(); }
    const float mu = SIGMA * red[0] / (float)NZ;
    __syncthreads();

    // ---- residual vectors & condensed rhs ----
    for (int idx = tid; idx < NZ; idx += NTH) {
      const int i = idx / NW, w = idx % NW;
      float gz = z[idx];                     // + I z
      const float* Krow = Km + (size_t)i * 128;
      for (int j = 0; j < NS; ++j) gz += Krow[j] * z[j * NW + w];
      const float yh = (labels[i] == w) ? 1.f : 0.f;
      const float r1v = gz - yh + lam[idx] + nu[i];
      const float r2v = z[idx] + sv[idx] - C_REG * yh;
      const float r3v = lam[idx] * sv[idx] - mu;
      r2[idx] = r2v; r3[idx] = r3v;
      rhs[idx] = -(r1v + (lam[idx] * r2v - r3v) / sv[idx]);
    }
    for (int i = tid; i < NP; i += NTH) {
      float r4 = 0.f;
      for (int w = 0; w < NW; ++w) r4 += z[i * NW + w];
      rhs[NZ + i] = -r4;
    }
    __syncthreads();

    // ---- build KKT = [[H, A^T], [A, 0]] ----
    for (int e = tid; e < NT * NT; e += NTH) {
      const int rr = e / NT, c = e % NT;
      float v = 0.f;
      if (rr < NZ && c < NZ) {
        const int i = rr / NW, w = rr % NW, j = c / NW, w2 = c % NW;
        if (w == w2) v = Km[(size_t)i * 128 + j];
        if (rr == c) v += 1.f + lam[rr] / sv[rr];
      } else if (rr < NZ) {          // A^T block
        v = (rr / NW == (c - NZ)) ? 1.f : 0.f;
      } else if (c < NZ) {           // A block
        v = (c / NW == (rr - NZ)) ? 1.f : 0.f;
      }
      KKT[(size_t)rr * KKTS + c] = v;
    }
    __syncthreads();

    // ---- LU with partial pivoting, rhs carried along ----
    for (int k = 0; k < NT; ++k) {
      if (tid == 0) {
        int pr = k; float best = fabsf(KKT[(size_t)k * KKTS + k]);
        for (int rr = k + 1; rr < NT; ++rr) {
          float a = fabsf(KKT[(size_t)rr * KKTS + k]);
          if (a > best) { best = a; pr = rr; }
        }
        s_piv = pr;
      }
      __syncthreads();
      const int pr = s_piv;
      if (pr != k) {
        for (int c = k + tid; c < NT; c += NTH) {
          float t = KKT[(size_t)k * KKTS + c];
          KKT[(size_t)k * KKTS + c] = KKT[(size_t)pr * KKTS + c];
          KKT[(size_t)pr * KKTS + c] = t;
        }
        if (tid == 0) { float t = rhs[k]; rhs[k] = rhs[pr]; rhs[pr] = t; }
      }
      __syncthreads();
      const float pivot = KKT[(size_t)k * KKTS + k];
      const float rhsk = rhs[k];
      const float* kr = KKT + (size_t)k * KKTS;
      for (int rr = k + 1 + tid; rr < NT; rr += NTH) {
        float* rowp = KKT + (size_t)rr * KKTS;
        const float f = rowp[k] / pivot;
        if (f != 0.f) {
          for (int c = k + 1; c < NT; ++c) rowp[c] -= f * kr[c];
          rhs[rr] -= f * rhsk;
        }
        rowp[k] = 0.f;
      }
      __syncthreads();
    }

    // ---- back substitution ----
    for (int k = NT - 1; k >= 0; --k) {
      float p = 0.f;
      const float* kr = KKT + (size_t)k * KKTS;
      for (int c = k + 1 + tid; c < NT; c += NTH) p += kr[c] * sol[c];
      red[tid] = p; __syncthreads();
      for (int st = 128; st > 0; st >>= 1) { if (tid < st) red[tid] += red[tid + st]; __syncthreads(); }
      if (tid == 0) sol[k] = (rhs[k] - red[0]) / kr[k];
      __syncthreads();
    }

    // ---- steps + fraction-to-boundary ----
    float amin = 1e30f;
    for (int idx = tid; idx < NZ; idx += NTH) {
      const float dz = sol[idx];
      const float dsx = -r2[idx] - dz;
      const float dlx = (-r3[idx] - lam[idx] * dsx) / sv[idx];
      dsv[idx] = dsx; dlam[idx] = dlx;
      if (dsx < 0.f) amin = fminf(amin, -sv[idx] / dsx);
      if (dlx < 0.f) amin = fminf(amin, -lam[idx] / dlx);
    }
    red[tid] = amin; __syncthreads();
    for (int st = 128; st > 0; st >>= 1) { if (tid < st) red[tid] = fminf(red[tid], red[tid + st]); __syncthreads(); }
    const float alpha = fminf(1.f, 0.99f * red[0]);
    __syncthreads();

    for (int idx = tid; idx < NZ; idx += NTH) {
      z[idx]   += alpha * sol[idx];
      sv[idx]  += alpha * dsv[idx];
      lam[idx] += alpha * dlam[idx];
    }
    for (int i = tid; i < NP; i += NTH) nu[i] += alpha * sol[NZ + i];
    __syncthreads();
  }
}

// ======================================================================
// Kernel 4: logits[q][w] = scale * sum_s z[s*5+w] * compat[s][q]
// ======================================================================
__global__ void logits_kernel(const float* __restrict__ ws,
                              const float* __restrict__ scale,
                              float* __restrict__ out) {
  const int idx = blockIdx.x * blockDim.x + threadIdx.x;
  if (idx >= NQ * NW) return;
  const int q = idx / NW, w = idx % NW;
  const float* z = ws + OFF_Z;
  const float* C = ws + OFF_COMPAT;
  float acc = 0.f;
  for (int s = 0; s < NS; ++s) acc += z[s * NW + w] * C[(size_t)s * NQ + q];
  out[(size_t)q * NW + w] = scale[0] * acc;
}

// ======================================================================
// Kernel 5: num_sv = count of support rows with any(z > 0.001)
// ======================================================================
__global__ void numsv_kernel(const float* __restrict__ ws, float* __restrict__ out,
                             int out_size) {
  __shared__ int red[128];
  const int tid = threadIdx.x;
  const float* z = ws + OFF_Z;
  int cnt = 0;
  for (int i = tid; i < NS; i += 128) {
    bool any = false;
    for (int w = 0; w < NW; ++w) any = any || (z[i * NW + w] > 0.001f);
    cnt += any ? 1 : 0;
  }
  red[tid] = cnt; __syncthreads();
  for (int st = 64; st > 0; st >>= 1) { if (tid < st) red[tid] += red[tid + st]; __syncthreads(); }
  if (tid == 0 && out_size > NQ * NW) out[NQ * NW] = (float)red[0];
}

// ======================================================================
extern "C" void kernel_launch(void* const* d_in, const int* in_sizes, int n_in,
                              void* d_out, int out_size, void* d_ws, size_t ws_size,
                              hipStream_t stream) {
  const float* query   = (const float*)d_in[0];   // (2048, 8192) f32
  const float* support = (const float*)d_in[1];   // (125, 8192) f32
  const int*   labels  = (const int*)d_in[2];     // (125,) i32
  // d_in[3] = n_way, d_in[4] = n_shot (compile-time constants here)
  const float* scale   = (const float*)d_in[5];   // (1,) f32

  float* ws  = (float*)d_ws;
  float* out = (float*)d_out;

  kmat_wmma<<<dim3(8, 8), 32, 0, stream>>>(support, ws);
  compat_wmma<<<dim3(NQ / 16, 8), 32, 0, stream>>>(support, query, ws);
  qp_solve_kernel<<<1, 256, 0, stream>>>(labels, ws);
  logits_kernel<<<(NQ * NW + 255) / 256, 256, 0, stream>>>(ws, scale, out);
  numsv_kernel<<<1, 128, 0, stream>>>(ws, out, out_size);
}